// GATLayer_8014408974363
// MI455X (gfx1250) — compile-verified
//
#include <hip/hip_runtime.h>
#include <hip/hip_bf16.h>

#define NN   50000
#define EE   1600000
#define DIN  128
#define HD   64
#define NH   4
#define DHEAD 16
#define SLOPE 0.2f
#define EPSV  1e-9f

typedef __attribute__((ext_vector_type(2))) float v2f;
typedef __attribute__((ext_vector_type(8))) float v8f;

// ---------------------------------------------------------------------------
// K0: fold attention vectors a1/a2/a3 into W, V, U -> small fused matrices.
// w_a[k][h] = sum_d W[k][h*16+d]*a[h][d]       (128x4)
// v_a[k][h] = sum_d V[k][h*16+d]*a[h][16+d]    (128x4)
// u_a[k][h] = sum_d U[k][h*16+d]*a[h][32+d]    (32x4)
// ---------------------------------------------------------------------------
__global__ __launch_bounds__(128) void fuse_weights(
    const float* __restrict__ W, const float* __restrict__ V,
    const float* __restrict__ U, const float* __restrict__ a,
    float* __restrict__ wa, float* __restrict__ va, float* __restrict__ ua)
{
    int k = threadIdx.x;  // 0..127
    for (int h = 0; h < NH; ++h) {
        float sw = 0.f, sv = 0.f;
        for (int d = 0; d < DHEAD; ++d) {
            sw += W[k * HD + h * DHEAD + d] * a[h * 48 + d];
            sv += V[k * HD + h * DHEAD + d] * a[h * 48 + 16 + d];
        }
        wa[k * NH + h] = sw;
        va[k * NH + h] = sv;
        if (k < 32) {
            float su = 0.f;
            for (int d = 0; d < DHEAD; ++d)
                su += U[k * HD + h * DHEAD + d] * a[h * 48 + 32 + d];
            ua[k * NH + h] = su;
        }
    }
}

// ---------------------------------------------------------------------------
// K1: zero the atomic-accumulated buffers (harness poisons them with 0xAA).
// ---------------------------------------------------------------------------
__global__ __launch_bounds__(256) void zero_init(float* __restrict__ out,
                                                 float* __restrict__ denom)
{
    int i = blockIdx.x * 256 + threadIdx.x;
    if (i < NN * HD) out[i] = 0.f;
    if (i < NN * NH) denom[i] = 0.f;
}

// ---------------------------------------------------------------------------
// K2: node GEMM via V_WMMA_F32_16X16X4_F32 (exact fp32, matches reference).
//   Vh    = X @ V                (N x 64)
//   s_src = X @ w_a, s_dst = X @ v_a   (N x 4 each, packed into 5th tile)
// Block = 320 threads = 10 waves; 32 node rows per block.
//   wave = rg*5 + ct : rg in {0,1} row-group of 16, ct in {0..4} col tile.
//   ct<4  -> Vh columns ct*16..+15 ; ct==4 -> [w_a | v_a | zero] tile.
// WMMA f32 16x16x4 per-lane layout (ISA 7.12.2):
//   A (16x4):  lane<16: M=lane, {K0,K1}; lane>=16: M=lane-16, {K2,K3}
//   B (4x16):  lane<16: N=lane, {K0,K1}; lane>=16: N=lane-16, {K2,K3}
//   C/D: vgpr r, lane<16 -> M=r,N=lane ; lane>=16 -> M=r+8,N=lane-16
// ---------------------------------------------------------------------------
__global__ __launch_bounds__(320) void node_gemm(
    const float* __restrict__ X, const float* __restrict__ V,
    const float* __restrict__ wa, const float* __restrict__ va,
    float* __restrict__ Vh, float* __restrict__ s_src, float* __restrict__ s_dst)
{
    __shared__ float sX[32][DIN + 1];   // +1 pad: avoid 16-way bank conflicts
    __shared__ float sV[DIN][HD + 1];   // +1 pad
    __shared__ float sB[DIN][16];       // [w_a | v_a | 0] padded tile

    const int tid = threadIdx.x;
    const int node_base = blockIdx.x * 32;

    // stage X tile (rows clamped so every lane has a valid address)
    for (int i = tid; i < 32 * DIN; i += 320) {
        int r = i >> 7, k = i & (DIN - 1);
        int gr = node_base + r; if (gr >= NN) gr = NN - 1;
        sX[r][k] = X[(size_t)gr * DIN + k];
    }
    // stage V (128x64, 32 KB resident -> cheap L2 re-reads)
    for (int i = tid; i < DIN * HD; i += 320) {
        int k = i >> 6, c = i & (HD - 1);
        sV[k][c] = V[i];
    }
    // stage fused-attention tile: cols 0-3 = w_a, 4-7 = v_a, 8-15 = 0
    for (int i = tid; i < DIN * 16; i += 320) {
        int k = i >> 4, c = i & 15;
        float v = 0.f;
        if (c < 4)      v = wa[k * NH + c];
        else if (c < 8) v = va[k * NH + (c - 4)];
        sB[k][c] = v;
    }
    __syncthreads();

    const int wave = tid >> 5;
    const int lane = tid & 31;
    const int half = lane >> 4;
    const int l    = lane & 15;
    const int rg   = wave / 5;   // 0..1
    const int ct   = wave % 5;   // 0..4
    const int row  = rg * 16 + l;

    const float* bp;
    int bs;
    if (ct < 4) { bp = &sV[0][ct * 16 + l]; bs = HD + 1; }
    else        { bp = &sB[0][l];           bs = 16;     }

    v8f acc = {};
    #pragma unroll 4
    for (int kc = 0; kc < DIN / 4; ++kc) {
        const int k0 = kc * 4 + half * 2;
        v2f a, b;
        a.x = sX[row][k0];
        a.y = sX[row][k0 + 1];
        b.x = bp[k0 * bs];
        b.y = bp[(k0 + 1) * bs];
        acc = __builtin_amdgcn_wmma_f32_16x16x4_f32(
            /*neg_a=*/false, a, /*neg_b=*/false, b,
            /*c_mod=*/(short)0, acc, /*reuse_a=*/false, /*reuse_b=*/false);
    }

    #pragma unroll
    for (int r = 0; r < 8; ++r) {
        const int m = r + (half ? 8 : 0);
        const int gnode = node_base + rg * 16 + m;
        if (gnode < NN) {
            if (ct < 4)      Vh[(size_t)gnode * HD + ct * 16 + l] = acc[r];
            else if (l < 4)  s_src[gnode * NH + l]       = acc[r];
            else if (l < 8)  s_dst[gnode * NH + (l - 4)] = acc[r];
        }
    }
}

// ---------------------------------------------------------------------------
// K3: per-edge logits. s_edge computed on the fly as edge_attr[e]·u_a (32x4
// folded matrix, LDS broadcast); leaky-ReLU; exp; atomic denom at dst.
// Streams edge_attr (204.8 MB) with float4 loads -> memory bound.
// ---------------------------------------------------------------------------
__global__ __launch_bounds__(256) void edge_logits(
    const float* __restrict__ EA, const int* __restrict__ src,
    const int* __restrict__ dst, const float* __restrict__ ua,
    const float* __restrict__ ssrc, const float* __restrict__ sdst,
    float* __restrict__ ex, float* __restrict__ denom)
{
    __shared__ float sU[32][NH];
    if (threadIdx.x < 128) sU[threadIdx.x >> 2][threadIdx.x & 3] = ua[threadIdx.x];
    __syncthreads();

    const int e = blockIdx.x * 256 + threadIdx.x;
    if (e >= EE) return;

    float accv[NH] = {0.f, 0.f, 0.f, 0.f};
    const float4* ea4 = (const float4*)(EA + (size_t)e * 32);
    #pragma unroll
    for (int q = 0; q < 8; ++q) {
        const float4 v = ea4[q];
        const int k = q * 4;
        #pragma unroll
        for (int h = 0; h < NH; ++h)
            accv[h] += v.x * sU[k][h] + v.y * sU[k + 1][h] +
                       v.z * sU[k + 2][h] + v.w * sU[k + 3][h];
    }
    const int s = src[e], d = dst[e];
    #pragma unroll
    for (int h = 0; h < NH; ++h) {
        float lg = ssrc[s * NH + h] + sdst[d * NH + h] + accv[h];
        lg = (lg >= 0.f) ? lg : SLOPE * lg;
        const float ev = expf(lg);
        ex[(size_t)e * NH + h] = ev;
        unsafeAtomicAdd(&denom[d * NH + h], ev);
    }
}

// ---------------------------------------------------------------------------
// K4: normalize + scatter. 8 threads per edge, each owns 8 contiguous
// components (one half-head): out[src] += Vh[dst] * alpha. Vh (12.8 MB) and
// out (12.8 MB) are L2-resident, so gathers/atomics stay on-chip.
// ---------------------------------------------------------------------------
__global__ __launch_bounds__(256) void scatter_msgs(
    const int* __restrict__ src, const int* __restrict__ dst,
    const float* __restrict__ ex, const float* __restrict__ denom,
    const float* __restrict__ Vh, float* __restrict__ out)
{
    const int gt = blockIdx.x * 256 + threadIdx.x;
    const int e  = gt >> 3;
    const int t8 = gt & 7;
    if (e >= EE) return;

    const int s = src[e], d = dst[e];
    const int h = t8 >> 1;
    const float alpha = ex[(size_t)e * NH + h] / (denom[d * NH + h] + EPSV);

    const float4* vp = (const float4*)(Vh + (size_t)d * HD + t8 * 8);
    const float4 v0 = vp[0], v1 = vp[1];
    float* op = out + (size_t)s * HD + t8 * 8;
    unsafeAtomicAdd(op + 0, v0.x * alpha);
    unsafeAtomicAdd(op + 1, v0.y * alpha);
    unsafeAtomicAdd(op + 2, v0.z * alpha);
    unsafeAtomicAdd(op + 3, v0.w * alpha);
    unsafeAtomicAdd(op + 4, v1.x * alpha);
    unsafeAtomicAdd(op + 5, v1.y * alpha);
    unsafeAtomicAdd(op + 6, v1.z * alpha);
    unsafeAtomicAdd(op + 7, v1.w * alpha);
}

// ---------------------------------------------------------------------------
extern "C" void kernel_launch(void* const* d_in, const int* in_sizes, int n_in,
                              void* d_out, int out_size, void* d_ws, size_t ws_size,
                              hipStream_t stream)
{
    const float* X  = (const float*)d_in[0];
    const float* EA = (const float*)d_in[1];
    const float* W  = (const float*)d_in[2];
    const float* V  = (const float*)d_in[3];
    const float* U  = (const float*)d_in[4];
    const float* a  = (const float*)d_in[5];
    const int* eidx = (const int*)d_in[6];
    const int* src  = eidx;        // edge_index[0]
    const int* dst  = eidx + EE;   // edge_index[1]
    float* out = (float*)d_out;

    // workspace layout (floats): ~40.8 MB total
    float* ws    = (float*)d_ws;
    float* Vh    = ws;                          // N*64
    float* s_src = Vh    + (size_t)NN * HD;     // N*4
    float* s_dst = s_src + (size_t)NN * NH;     // N*4
    float* denom = s_dst + (size_t)NN * NH;     // N*4
    float* ex    = denom + (size_t)NN * NH;     // E*4
    float* wa    = ex    + (size_t)EE * NH;     // 512
    float* va    = wa + DIN * NH;               // 512
    float* ua    = va + DIN * NH;               // 128

    fuse_weights<<<1, 128, 0, stream>>>(W, V, U, a, wa, va, ua);
    zero_init<<<(NN * HD + 255) / 256, 256, 0, stream>>>(out, denom);
    node_gemm<<<(NN + 31) / 32, 320, 0, stream>>>(X, V, wa, va, Vh, s_src, s_dst);
    edge_logits<<<(EE + 255) / 256, 256, 0, stream>>>(EA, src, dst, ua, s_src,
                                                      s_dst, ex, denom);
    scatter_msgs<<<((size_t)EE * 8 + 255) / 256, 256, 0, stream>>>(src, dst, ex,
                                                                   denom, Vh, out);
}